// UniformLaplacian_39814346834440
// MI455X (gfx1250) — compile-verified
//
#include <hip/hip_runtime.h>

// ---------------------------------------------------------------------------
// Uniform mesh Laplacian: x_i = (deg_i * V_i - sum_{j in N(i)} V_j)/(deg_i+eps)
// verts: [B=4, N, 64] f32, faces: [B, F, 3] i64, out: [B, N, 64] f32
// Per face, each vertex receives the SUM of the other two rows, deg += 2.
// L2-resident f32 atomics (192MB L2 >> ~112MB working set). Face indices are
// staged per-block into LDS with GLOBAL_LOAD_ASYNC_TO_LDS_B128 (ASYNCcnt).
// ---------------------------------------------------------------------------

#define FACES_PER_BLOCK 32                 // 8 waves x 4 faces
#define TILE_BYTES      (FACES_PER_BLOCK * 3 * 8)   // 768 B
#define TILE_CHUNKS     (TILE_BYTES / 16)           // 48 x b128

__global__ __launch_bounds__(256) void ul_zero_f4(float4* __restrict__ p, int n4) {
    int i = blockIdx.x * blockDim.x + threadIdx.x;
    if (i < n4) p[i] = make_float4(0.f, 0.f, 0.f, 0.f);
}

__global__ __launch_bounds__(256) void ul_scatter(
    const float* __restrict__ verts,        // [B*N, 64]
    const long long* __restrict__ faces,    // [B*F, 3]
    float* __restrict__ accum,              // [B*N, 64]  (= d_out)
    float* __restrict__ deg,                // [B*N]
    int F, int N, int totalFaces) {

    __shared__ long long sF[FACES_PER_BLOCK * 3];

    const int tid        = threadIdx.x;
    const int blockFace0 = blockIdx.x * FACES_PER_BLOCK;
    const size_t facesBytes = (size_t)totalFaces * 24;

    // ---- async stage: copy this block's 768B face tile into LDS -----------
    {
        size_t srcByte = (size_t)blockFace0 * 24 + (size_t)tid * 16;
        if (tid < TILE_CHUNKS && srcByte + 16 <= facesBytes) {
            unsigned lds = (unsigned)(uintptr_t)sF + (unsigned)(tid * 16);
            unsigned long long ga =
                (unsigned long long)(const char*)faces + srcByte;
            asm volatile("global_load_async_to_lds_b128 %0, %1, off"
                         :: "v"(lds), "v"(ga) : "memory");
        }
        // Issuing waves drain their ASYNCcnt; others no-op (cnt==0).
        asm volatile("s_wait_asynccnt 0x0" ::: "memory");
    }

    // Stream-prefetch the face tile ~64 blocks ahead (global_prefetch_b8,
    // one cacheline per lane).
    if (tid < TILE_BYTES / 128) {
        size_t ahead = ((size_t)blockIdx.x + 64) * TILE_BYTES + (size_t)tid * 128;
        if (ahead + 128 <= facesBytes)
            __builtin_prefetch((const char*)faces + ahead, 0, 0);
    }

    __syncthreads();

    const int lane = tid & 31;
    const int waveInBlk = tid >> 5;         // 0..7, wave-uniform
    const int* sF32 = (const int*)sF;       // low dword of each i64 index

    for (int it = 0; it < FACES_PER_BLOCK / 8; ++it) {
        // local face id; force into SGPR (it is wave-uniform by construction)
        int lf   = __builtin_amdgcn_readfirstlane(waveInBlk + it * 8);
        int face = blockFace0 + lf;
        if (face >= totalFaces) break;      // wave-uniform

        int b    = face / F;                // scalar
        int base = b * N;                   // block-diagonal batch offset

        // indices fit in 32 bits (B*N = 200k): read low dword from LDS
        int i0 = __builtin_amdgcn_readfirstlane(sF32[(lf * 3 + 0) * 2]) + base;
        int i1 = __builtin_amdgcn_readfirstlane(sF32[(lf * 3 + 1) * 2]) + base;
        int i2 = __builtin_amdgcn_readfirstlane(sF32[(lf * 3 + 2) * 2]) + base;

        // Each lane handles channels {2*lane, 2*lane+1}: one 8B load per row.
        const float2* V2 = (const float2*)verts;
        float2 a  = V2[(size_t)i0 * 32 + lane];
        float2 c1 = V2[(size_t)i1 * 32 + lane];
        float2 c2 = V2[(size_t)i2 * 32 + lane];

        float2 s0 = make_float2(c1.x + c2.x, c1.y + c2.y);  // -> vertex i0
        float2 s1 = make_float2(a.x  + c2.x, a.y  + c2.y);  // -> vertex i1
        float2 s2 = make_float2(a.x  + c1.x, a.y  + c1.y);  // -> vertex i2

        float* p0 = accum + ((size_t)i0 << 6) + lane * 2;
        float* p1 = accum + ((size_t)i1 << 6) + lane * 2;
        float* p2 = accum + ((size_t)i2 << 6) + lane * 2;

        // Hardware L2 f32 atomic adds (no return -> STOREcnt path).
        unsafeAtomicAdd(p0,     s0.x);
        unsafeAtomicAdd(p0 + 1, s0.y);
        unsafeAtomicAdd(p1,     s1.x);
        unsafeAtomicAdd(p1 + 1, s1.y);
        unsafeAtomicAdd(p2,     s2.x);
        unsafeAtomicAdd(p2 + 1, s2.y);

        // Each face contributes 2 half-edges to each of its 3 vertices.
        if (lane < 3) {
            int t = (lane == 0) ? i0 : (lane == 1) ? i1 : i2;
            unsafeAtomicAdd(deg + t, 2.0f);
        }
    }
}

__global__ __launch_bounds__(256) void ul_finalize(
    const float* __restrict__ verts,
    const float* __restrict__ deg,
    float* __restrict__ out,      // holds neigh on entry, x on exit
    int nThreads) {               // = B*N*16 (one float4 per thread)

    int t = blockIdx.x * blockDim.x + threadIdx.x;
    if (t >= nThreads) return;
    int v = t >> 4;               // vertex row (16 float4 per 64-ch row)
    float d   = deg[v];
    float inv = 1.0f / (d + 1e-12f);

    float4 V = ((const float4*)verts)[t];
    float4 n = ((float4*)out)[t];
    float4 x;
    x.x = (d * V.x - n.x) * inv;
    x.y = (d * V.y - n.y) * inv;
    x.z = (d * V.z - n.z) * inv;
    x.w = (d * V.w - n.w) * inv;
    ((float4*)out)[t] = x;
}

extern "C" void kernel_launch(void* const* d_in, const int* in_sizes, int n_in,
                              void* d_out, int out_size, void* d_ws, size_t ws_size,
                              hipStream_t stream) {
    const float*     verts = (const float*)d_in[0];
    const long long* faces = (const long long*)d_in[1];
    float* out = (float*)d_out;
    float* deg = (float*)d_ws;

    const int totalVF    = in_sizes[0];        // B*N*64 = 12,800,000
    const int BN         = totalVF / 64;       // B*N    = 200,000
    const int totalFaces = in_sizes[1] / 3;    // B*F    = 400,000
    const int B = 4;
    const int N = BN / B;
    const int F = totalFaces / B;

    // 1) zero accumulator (d_out) and degree scratch
    {
        int n4 = totalVF / 4;
        ul_zero_f4<<<(n4 + 255) / 256, 256, 0, stream>>>((float4*)out, n4);
        int n4d = BN / 4;                       // BN divisible by 4
        ul_zero_f4<<<(n4d + 255) / 256, 256, 0, stream>>>((float4*)deg, n4d);
    }

    // 2) scatter: one wave32 per face, 32 faces per block, LDS-staged indices
    {
        int blocks = (totalFaces + FACES_PER_BLOCK - 1) / FACES_PER_BLOCK;
        ul_scatter<<<blocks, 256, 0, stream>>>(verts, faces, out, deg,
                                               F, N, totalFaces);
    }

    // 3) finalize: x = (deg*V - neigh)/(deg+eps), float4-vectorized in place
    {
        int threads = BN * 16;
        ul_finalize<<<(threads + 255) / 256, 256, 0, stream>>>(verts, deg, out,
                                                               threads);
    }
}